// AdaptivePrototypes_75746043232674
// MI455X (gfx1250) — compile-verified
//
#include <hip/hip_runtime.h>
#include <math.h>

typedef __attribute__((ext_vector_type(2))) float v2f;
typedef __attribute__((ext_vector_type(8))) float v8f;

#define NPROTO 64
#define PDIM 512
#define NFEAT 200000
#define KSTEPS (PDIM / 4)        // 128 WMMA K-steps
#define NTILE (NPROTO / 16)      // 4 N-tiles of 16 prototypes
#define ROWTILES (NFEAT / 16)    // 12500 row tiles of 16 rows
#define WAVES_PER_BLOCK 8
#define BLOCK_THREADS 256
#define MAX_PARTIALS 256
#define PARTIAL_FLOATS (NPROTO * PDIM + NPROTO)  // 32832: sums + counts
#define UPDATE_RATE 0.1f

// ---------------------------------------------------------------------------
// Kernel 1: normalize prototypes and store them in WMMA B-fragment layout.
// B for V_WMMA_F32_16X16X4_F32 (4x16, K x N), 2 VGPRs:
//   lane L, vgpr j  ->  B[k = j + 2*(L/16)][n = L%16]
// For N-tile t, k-step s:  value = p_hat[16t + L%16][4s + (L/16)*2 + j]
// Stored at bfrag[((s*4 + t)*32 + lane)*2 + j]  -> hot-loop loads are
// contiguous 256B v2f loads per (s,t).
// ---------------------------------------------------------------------------
__global__ void proto_prep(const float* __restrict__ protos,
                           float* __restrict__ bfrag) {
  __shared__ float red[256];
  const int p = blockIdx.x;      // prototype row, 64 blocks
  const int tid = threadIdx.x;   // 256 threads
  const float x0 = protos[p * PDIM + tid];
  const float x1 = protos[p * PDIM + tid + 256];
  red[tid] = x0 * x0 + x1 * x1;
  __syncthreads();
  for (int off = 128; off > 0; off >>= 1) {
    if (tid < off) red[tid] += red[tid + off];
    __syncthreads();
  }
  const float inv = 1.0f / fmaxf(sqrtf(red[0]), 1e-8f);
  const int t = p >> 4;
  const int lm = p & 15;
#pragma unroll
  for (int e = 0; e < 2; ++e) {
    const int d = tid + e * 256;
    const float val = (e ? x1 : x0) * inv;
    const int s = d >> 2;
    const int r = d & 3;
    const int lane = lm + ((r >> 1) << 4);
    const int j = r & 1;
    bfrag[((s * NTILE + t) * 32 + lane) * 2 + j] = val;
  }
}

// ---------------------------------------------------------------------------
// Kernel 2: persistent. Per wave: 16-row tile -> sims via WMMA f32 16x16x4,
// argmax via shfl_xor reduction, then LDS ds_add_f32 scatter of raw rows.
// Block-level 128KB LDS partial flushed to workspace (no global atomics).
// ---------------------------------------------------------------------------
__global__ void __launch_bounds__(BLOCK_THREADS) assign_accum(
    const float* __restrict__ features, const float* __restrict__ bfrag,
    float* __restrict__ partials, int nPartial) {
  extern __shared__ float lds[];  // [NPROTO*PDIM sums][NPROTO counts]
  float* s_sums = lds;
  float* s_counts = lds + NPROTO * PDIM;

  for (int i = threadIdx.x; i < PARTIAL_FLOATS; i += BLOCK_THREADS)
    lds[i] = 0.0f;
  __syncthreads();

  const int lane = threadIdx.x & 31;
  const int wave = threadIdx.x >> 5;
  const int gw = blockIdx.x * WAVES_PER_BLOCK + wave;
  const int nw = gridDim.x * WAVES_PER_BLOCK;
  const int halfSel = lane >> 4;  // 0: lanes 0-15, 1: lanes 16-31
  const int lm = lane & 15;
  const v2f* __restrict__ bPtr = (const v2f*)bfrag;

  for (int tile = gw; tile < ROWTILES; tile += nw) {
    const int rowBase = tile * 16;
    // A fragment base (16x4 f32 layout): lane L reads row rowBase+L%16,
    // columns 4s + (L/16)*2 .. +1  -> one v2f load per step.
    const float* aBase = features + (size_t)(rowBase + lm) * PDIM + halfSel * 2;

    v8f c0 = {}, c1 = {}, c2 = {}, c3 = {};
#pragma unroll 4
    for (int s = 0; s < KSTEPS; ++s) {
      const v2f a = *(const v2f*)(aBase + s * 4);
      const v2f b0 = bPtr[(s * NTILE + 0) * 32 + lane];
      const v2f b1 = bPtr[(s * NTILE + 1) * 32 + lane];
      const v2f b2 = bPtr[(s * NTILE + 2) * 32 + lane];
      const v2f b3 = bPtr[(s * NTILE + 3) * 32 + lane];
      c0 = __builtin_amdgcn_wmma_f32_16x16x4_f32(false, a, false, b0, (short)0, c0, false, false);
      c1 = __builtin_amdgcn_wmma_f32_16x16x4_f32(false, a, false, b1, (short)0, c1, false, false);
      c2 = __builtin_amdgcn_wmma_f32_16x16x4_f32(false, a, false, b2, (short)0, c2, false, false);
      c3 = __builtin_amdgcn_wmma_f32_16x16x4_f32(false, a, false, b3, (short)0, c3, false, false);
    }

    // Argmax per row. C layout: vgpr v, lane L -> row v + 8*(L/16),
    // proto 16t + L%16. Per-lane best over 4 tiles, then reduce over the
    // 16 lanes sharing the row (xor masks 1..8 stay inside each half).
    int assign[8];
#pragma unroll
    for (int v = 0; v < 8; ++v) {
      float bestV = c0[v];
      int bestI = lm;
      const float t1 = c1[v], t2 = c2[v], t3 = c3[v];
      if (t1 > bestV) { bestV = t1; bestI = lm + 16; }
      if (t2 > bestV) { bestV = t2; bestI = lm + 32; }
      if (t3 > bestV) { bestV = t3; bestI = lm + 48; }
#pragma unroll
      for (int m = 8; m >= 1; m >>= 1) {
        const float ov = __shfl_xor(bestV, m, 32);
        const int oi = __shfl_xor(bestI, m, 32);
        if (ov > bestV || (ov == bestV && oi < bestI)) { bestV = ov; bestI = oi; }
      }
      assign[v] = bestI;  // all 16 lanes of the half agree
    }

    // Counts: one lane per half owns its 8 rows.
    if (lm == 0) {
#pragma unroll
      for (int v = 0; v < 8; ++v) atomicAdd(&s_counts[assign[v]], 1.0f);
    }

    // Scatter raw feature rows (WGP$-hot re-read) into LDS partial sums.
    // Row r: assignment lives in lane (r/8)*16, register assign[r%8].
#pragma unroll
    for (int r = 0; r < 16; ++r) {
      const int a = __shfl(assign[r & 7], (r >> 3) << 4, 32);
      const float* frow = features + (size_t)(rowBase + r) * PDIM;
      float* srow = s_sums + a * PDIM;
#pragma unroll
      for (int jj = 0; jj < 4; ++jj) {
        const int d = jj * 128 + lane * 4;
        const float4 x = *(const float4*)(frow + d);
        atomicAdd(&srow[d + 0], x.x);
        atomicAdd(&srow[d + 1], x.y);
        atomicAdd(&srow[d + 2], x.z);
        atomicAdd(&srow[d + 3], x.w);
      }
    }
  }

  __syncthreads();
  float* out = partials + (size_t)blockIdx.x * PARTIAL_FLOATS;
  for (int i = threadIdx.x; i < PARTIAL_FLOATS; i += BLOCK_THREADS)
    out[i] = lds[i];
}

// ---------------------------------------------------------------------------
// Kernel 3: reduce partials, compute means, EMA update, masked write.
// ---------------------------------------------------------------------------
__global__ void reduce_update(const float* __restrict__ partials, int nPartial,
                              const float* __restrict__ protos,
                              float* __restrict__ out) {
  __shared__ float red[256];
  const int k = blockIdx.x;     // 64 blocks, one prototype each
  const int tid = threadIdx.x;  // 256 threads

  float c = 0.0f;
  for (int p = tid; p < nPartial; p += 256)
    c += partials[(size_t)p * PARTIAL_FLOATS + NPROTO * PDIM + k];
  red[tid] = c;
  __syncthreads();
  for (int off = 128; off > 0; off >>= 1) {
    if (tid < off) red[tid] += red[tid + off];
    __syncthreads();
  }
  const float cnt = red[0];

  float s0 = 0.0f, s1 = 0.0f;
  for (int p = 0; p < nPartial; ++p) {
    const float* base = partials + (size_t)p * PARTIAL_FLOATS + k * PDIM;
    s0 += base[tid];
    s1 += base[tid + 256];
  }
  const float inv = 1.0f / fmaxf(cnt, 1.0f);
#pragma unroll
  for (int e = 0; e < 2; ++e) {
    const int d = tid + e * 256;
    const float pr = protos[k * PDIM + d];
    const float mean = (e ? s1 : s0) * inv;
    const float up = (1.0f - UPDATE_RATE) * pr + UPDATE_RATE * mean;
    out[k * PDIM + d] = (cnt > 0.0f) ? up : pr;
  }
}

extern "C" void kernel_launch(void* const* d_in, const int* in_sizes, int n_in,
                              void* d_out, int out_size, void* d_ws,
                              size_t ws_size, hipStream_t stream) {
  (void)in_sizes; (void)n_in; (void)out_size;
  const float* features = (const float*)d_in[0];
  const float* protos = (const float*)d_in[1];
  float* out = (float*)d_out;

  float* bfrag = (float*)d_ws;                       // 32768 floats (128 KB)
  const size_t bfragFloats = (size_t)NPROTO * PDIM;
  float* partials = bfrag + bfragFloats;

  // Size the persistent grid by available workspace (cap at MAX_PARTIALS).
  long cap = ((long)(ws_size / 4) - (long)bfragFloats) / PARTIAL_FLOATS;
  int nPartial = (int)(cap < 1 ? 1 : (cap > MAX_PARTIALS ? MAX_PARTIALS : cap));

  proto_prep<<<NPROTO, 256, 0, stream>>>(protos, bfrag);

  const size_t ldsBytes = (size_t)PARTIAL_FLOATS * sizeof(float);  // ~128.25 KB
  assign_accum<<<nPartial, BLOCK_THREADS, ldsBytes, stream>>>(
      features, bfrag, partials, nPartial);

  reduce_update<<<NPROTO, 256, 0, stream>>>(partials, nPartial, protos, out);
}